// PaiIndexMatrix_41154376631100
// MI455X (gfx1250) — compile-verified
//
#include <hip/hip_runtime.h>
#include <hip/hip_bf16.h>

// ---------------- problem constants (from setup_inputs) ----------------
#define B_   8
#define N_   4096
#define KNN  32          // k neighbors
#define KS   32          // kernel_size
#define NQ   16          // query points per workgroup (one WMMA M-tile)
#define CH   512         // candidate chunk held in LDS
#define NCHUNK (N_ / CH) // 8
#define TPB  256         // 8 wave32 waves

typedef __attribute__((ext_vector_type(2))) float v2f;
typedef __attribute__((ext_vector_type(8))) float v8f;

__global__ __launch_bounds__(TPB) void pai_knn_adj_kernel(
    const float* __restrict__ x,        // [B,3,N]
    const float* __restrict__ kernals,  // [3, KS-1]
    int*   __restrict__ out_idx,        // [B*N*KNN]
    float* __restrict__ out_w)          // [B*N, KNN, KS]
{
    __shared__ float pw[NQ * CH];              // 32 KB pairwise chunk
    __shared__ float topv[2][NQ * KNN];        // ping-pong top-k values
    __shared__ int   topi[2][NQ * KNN];        // ping-pong top-k indices
    __shared__ float rel[NQ * KNN * 3];        // gathered relative coords

    const int   tid  = threadIdx.x;
    const int   lane = tid & 31;
    const int   wave = tid >> 5;
    const int   hi   = lane >> 4;              // 0: lanes 0-15, 1: lanes 16-31
    const int   blk  = blockIdx.x;
    const int   b    = blk / (N_ / NQ);
    const int   q0   = (blk % (N_ / NQ)) * NQ;
    const float NEGINF = -__builtin_inff();
    const float* __restrict__ X = x + (size_t)b * 3 * N_;

    // init top-list buffer 0 (chunk 0 reads it as "old")
    for (int t = tid; t < NQ * KNN; t += TPB) {
        topv[0][t] = NEGINF;
        topi[0][t] = 0x7fffffff;
    }

    // ---- A-matrix (16 queries x 4, K=3 padded): per ISA 7.12.2 f32 16x4 layout
    const int qm = q0 + (lane & 15);
    const float qx = X[0 * N_ + qm];
    const float qy = X[1 * N_ + qm];
    const float qz = X[2 * N_ + qm];
    const float sqq = qx * qx + qy * qy + qz * qz;
    v2f amat;
    amat.x = (hi == 0) ? qx : qz;   // VGPR0: K=0 (lanes<16) / K=2 (lanes>=16)
    amat.y = (hi == 0) ? qy : 0.0f; // VGPR1: K=1 / K=3 (zero pad)
    float sqqrow[8];
    #pragma unroll
    for (int i = 0; i < 8; ++i)     // sq of the C-row this lane/VGPR sees
        sqqrow[i] = __shfl(sqq, i + hi * 8, 32);

    __syncthreads();

    for (int chunk = 0; chunk < NCHUNK; ++chunk) {
        const int cbase = chunk * CH;

        // ---------- phase 1: WMMA pairwise tile -> LDS ----------
        #pragma unroll
        for (int tt = 0; tt < CH / 16 / 8; ++tt) {   // 4 tiles per wave
            const int tile = wave * (CH / 16 / 8) + tt;
            const int c = cbase + tile * 16 + (lane & 15);
            const float cx = X[0 * N_ + c];
            const float cy = X[1 * N_ + c];
            const float cz = X[2 * N_ + c];
            const float sqc = cx * cx + cy * cy + cz * cz;
            v2f bmat;
            bmat.x = (hi == 0) ? cx : cz;
            bmat.y = (hi == 0) ? cy : 0.0f;
            v8f acc = {0.f, 0.f, 0.f, 0.f, 0.f, 0.f, 0.f, 0.f};
            // D = A(16x4) * B(4x16) + 0  -> inner products of query/candidate coords
            acc = __builtin_amdgcn_wmma_f32_16x16x4_f32(
                false, amat, false, bmat, (short)0, acc, false, false);
            const int nloc = tile * 16 + (lane & 15);
            #pragma unroll
            for (int i = 0; i < 8; ++i) {
                // pairwise = 2*inner - |q|^2 - |c|^2  (<= 0, max at nearest)
                const float v = 2.0f * acc[i] - sqqrow[i] - sqc;
                pw[(i + 8 * hi) * CH + nloc] = v;
            }
        }
        __syncthreads();

        // ---------- phase 2: streaming top-32 merge (2 rows per wave) ----------
        const int po = chunk & 1;   // old list parity
        const int pn = po ^ 1;      // new list parity
        for (int rr = 0; rr < 2; ++rr) {
            const int r = wave * 2 + rr;
            float* rowp = &pw[r * CH];
            const float* ov = &topv[po][r * KNN];
            const int*   oi = &topi[po][r * KNN];
            float* nv = &topv[pn][r * KNN];
            int*   ni = &topi[pn][r * KNN];
            int oldPos = 0;         // old list is sorted desc -> single head
            for (int s = 0; s < KNN; ++s) {
                float bv = NEGINF;
                int   bi = 0x7fffffff;
                #pragma unroll
                for (int u = 0; u < CH / 32; ++u) {
                    const int j = u * 32 + lane;
                    const float v = rowp[j];
                    const int  id = cbase + j;
                    if (v > bv || (v == bv && id < bi)) { bv = v; bi = id; }
                }
                if (lane == 0 && oldPos < KNN) {
                    const float v = ov[oldPos];
                    const int  id = oi[oldPos];
                    if (v > bv || (v == bv && id < bi)) { bv = v; bi = id; }
                }
                // wave32 argmax, stable tie-break on smaller index (JAX top_k)
                #pragma unroll
                for (int off = 16; off > 0; off >>= 1) {
                    const float tv = __shfl_xor(bv, off, 32);
                    const int   ti = __shfl_xor(bi, off, 32);
                    if (tv > bv || (tv == bv && ti < bi)) { bv = tv; bi = ti; }
                }
                if (bi < cbase) {
                    ++oldPos;                       // consumed old-list head
                } else if (lane == 0) {
                    rowp[bi - cbase] = NEGINF;      // mark chunk candidate
                }
                if (lane == 0) { nv[s] = bv; ni[s] = bi; }
            }
        }
        __syncthreads();
    }

    // final lists: last chunk (7, odd) wrote parity 0
    const int pf = ((NCHUNK - 1) & 1) ^ 1;
    const int* __restrict__ fi = topi[pf];

    // ---------- phase 3a: spirals_index + gather rel ----------
    for (int t = tid; t < NQ * KNN; t += TPB) {
        const int r = t >> 5, s = t & 31;
        const int nidx = fi[r * KNN + s];
        const int n0   = fi[r * KNN + 0];
        out_idx[((size_t)(b * N_ + q0 + r)) * KNN + s] = b * N_ + nidx;
        rel[t * 3 + 0] = X[0 * N_ + nidx] - X[0 * N_ + n0];
        rel[t * 3 + 1] = X[1 * N_ + nidx] - X[1 * N_ + n0];
        rel[t * 3 + 2] = X[2 * N_ + nidx] - X[2 * N_ + n0];
    }
    __syncthreads();

    // ---------- phase 3b: adjweight (one (point, kernel-col) column/thread) ----
    for (int t = tid; t < NQ * KS; t += TPB) {
        const int r = t >> 5, c = t & 31;
        float k0 = 0.f, k1 = 0.f, k2 = 0.f;
        if (c > 0) {                    // kernals_full col 0 is zeros
            k0 = kernals[0 * (KS - 1) + c - 1];
            k1 = kernals[1 * (KS - 1) + c - 1];
            k2 = kernals[2 * (KS - 1) + c - 1];
        }
        float w[KNN];
        float sum1 = 0.f;
        #pragma unroll
        for (int j = 0; j < KNN; ++j) {
            const float* rp = &rel[(r * KNN + j) * 3];
            float v = rp[0] * k0 + rp[1] * k1 + rp[2] * k2;
            if (j == 0 && c == 0) v += 1.0f;    // one_padding at [0,0]
            v = fmaxf(v, 0.0f);
            w[j] = v;
            sum1 += v;
        }
        const float d1 = sum1 + 1e-6f;
        float sum2 = 0.f;
        #pragma unroll
        for (int j = 0; j < KNN; ++j) {
            float v = w[j] / d1;
            v = v * v;
            w[j] = v;
            sum2 += v;
        }
        const float d2 = sum2 + 1e-6f;
        const size_t base = ((size_t)(b * N_ + q0 + r) * KNN) * KS + c;
        #pragma unroll
        for (int j = 0; j < KNN; ++j) {
            float v = w[j] / d2;
            v = (v > 0.1f) ? v : 0.0f;
            out_w[base + (size_t)j * KS] = v;   // coalesced: wave = 32 consecutive c
        }
    }
}

extern "C" void kernel_launch(void* const* d_in, const int* in_sizes, int n_in,
                              void* d_out, int out_size, void* d_ws, size_t ws_size,
                              hipStream_t stream) {
    (void)in_sizes; (void)n_in; (void)d_ws; (void)ws_size; (void)out_size;
    const float* x    = (const float*)d_in[0];   // [8,3,4096] fp32
    const float* kern = (const float*)d_in[1];   // [3,31] fp32
    // d_out = [spirals_index int32 (B*N*k)] ++ [adjweight fp32 (B*N*k*K)]
    int*   out_idx = (int*)d_out;
    float* out_w   = (float*)d_out + (size_t)B_ * N_ * KNN;

    dim3 grid(B_ * (N_ / NQ));   // 2048 workgroups
    pai_knn_adj_kernel<<<grid, TPB, 0, stream>>>(x, kern, out_idx, out_w);
}